// GNN_8134668059260
// MI455X (gfx1250) — compile-verified
//
#include <hip/hip_runtime.h>
#include <hip/hip_bf16.h>

typedef __attribute__((ext_vector_type(2))) float v2f;
typedef __attribute__((ext_vector_type(8))) float v8f;

#define N_NODES 50000
#define N_EDGES 800000
#define N_GRAPHS 128
#define IN_DIM 128
#define HID_DIM 96
#define OUT_DIM 10

// ---------------------------------------------------------------------------
// init: zero degree counts (int), pooled accumulators, graph counts
// ---------------------------------------------------------------------------
__global__ void k_init(int* __restrict__ deg, float* __restrict__ pooled,
                       float* __restrict__ cnt) {
    int t = blockIdx.x * blockDim.x + threadIdx.x;
    if (t < N_NODES) deg[t] = 0;
    if (t < N_GRAPHS * HID_DIM) pooled[t] = 0.0f;
    if (t < N_GRAPHS) cnt[t] = 0.0f;
}

// ---------------------------------------------------------------------------
// in-degree on destination (col) nodes
// ---------------------------------------------------------------------------
__global__ void k_degree(const int* __restrict__ col, int* __restrict__ deg) {
    int t = blockIdx.x * blockDim.x + threadIdx.x;
    if (t < N_EDGES) atomicAdd(&deg[col[t]], 1);
}

// dinv[i] = (1 + indeg[i])^-1/2   (aliases deg buffer: elementwise read->write)
__global__ void k_dinv(const int* __restrict__ deg, float* __restrict__ dinv) {
    int t = blockIdx.x * blockDim.x + threadIdx.x;
    if (t < N_NODES) {
        int d = deg[t];
        dinv[t] = rsqrtf(1.0f + (float)d);
    }
}

// ---------------------------------------------------------------------------
// WMMA GEMM: C[M x 96] = (DO_RELU ? relu(A) : A)[M x K] @ W[K x 96]
// One wave32 per 16-row M tile; 6 accumulator fragments cover N=96.
// V_WMMA_F32_16X16X4_F32 (exact fp32 path, matches reference precision).
// W is staged into LDS pair-interleaved so every B fragment is one
// ds_load_b64 straight into an even-aligned VGPR pair (no repacking).
// K is a compile-time constant -> K loop fully unrolls, letting the
// scheduler hoist global/LDS loads across WMMA groups instead of
// draining loadcnt/dscnt to zero every step.
// ---------------------------------------------------------------------------
template <int K, int DO_RELU>
__global__ void __launch_bounds__(256) k_gemm_wmma(
    const float* __restrict__ A, const float* __restrict__ W,
    float* __restrict__ C, int M, int numMTiles) {
    // pair-interleaved: sW[(r>>1)*192 + c*2 + (r&1)] = W[r*96 + c]
    __shared__ float sW[IN_DIM * HID_DIM]; // 48KB max case

    const int tid = threadIdx.x;
    for (int i = tid; i < K * HID_DIM; i += 256) {
        int r = i / HID_DIM;
        int c = i - r * HID_DIM;
        sW[(r >> 1) * (2 * HID_DIM) + c * 2 + (r & 1)] = W[i];
    }
    __syncthreads();

    const int wave = __builtin_amdgcn_readfirstlane(tid >> 5); // wave-uniform
    const int lane = tid & 31;
    const int mtile = blockIdx.x * 8 + wave;
    if (mtile >= numMTiles) return;

    const int lane16 = lane & 15;
    const int half = lane >> 4; // 0: K pair {kk,kk+1}, 1: K pair {kk+2,kk+3}
    const int m0 = mtile * 16;

    int mrow = m0 + lane16;
    if (mrow >= M) mrow = M - 1; // clamp (M is a multiple of 16 here anyway)
    const float* arow = A + (size_t)mrow * K;

    // B fragments viewed as v2f: element (pairRow, col) at sB[pairRow*96+col]
    const v2f* sB = (const v2f*)sW;
    const v2f* bBase = sB + (size_t)half * HID_DIM + lane16;

    v8f acc[6] = {}; // 6 * (16x16 f32 frag) = 16 rows x 96 cols

#pragma unroll
    for (int kk = 0; kk < K; kk += 4) {
        float2 av = *(const float2*)(arow + kk + 2 * half);
        if (DO_RELU) {
            av.x = fmaxf(av.x, 0.0f);
            av.y = fmaxf(av.y, 0.0f);
        }
        v2f a;
        a.x = av.x;
        a.y = av.y;

        const v2f* bp = bBase + (kk >> 1) * HID_DIM; // pair row kk/2 + half
#pragma unroll
        for (int n = 0; n < 6; ++n) {
            v2f b = bp[n * 16]; // single ds_load_b64
            acc[n] = __builtin_amdgcn_wmma_f32_16x16x4_f32(
                false, a, false, b, (short)0, acc[n], false, false);
        }
    }

#pragma unroll
    for (int n = 0; n < 6; ++n) {
        float* cp = C + (size_t)(m0 + 8 * half) * HID_DIM + n * 16 + lane16;
#pragma unroll
        for (int r = 0; r < 8; ++r) {
            int m = m0 + 8 * half + r;
            if (m < M) cp[(size_t)r * HID_DIM] = acc[n][r];
        }
    }
}

// ---------------------------------------------------------------------------
// agg[i][d..d+3] = dinv[i]^2 * h[i][d..d+3] + b[d..d+3]  (float4 vectorized;
// HID_DIM % 4 == 0 so a float4 never crosses a row boundary)
// ---------------------------------------------------------------------------
__global__ void k_selfloop(const float4* __restrict__ h,
                           const float* __restrict__ dinv,
                           const float4* __restrict__ b,
                           float4* __restrict__ agg) {
    const int nq = N_NODES * (HID_DIM / 4);
    int t = blockIdx.x * blockDim.x + threadIdx.x;
    if (t < nq) {
        int i = t / (HID_DIM / 4);
        int dq = t - i * (HID_DIM / 4);
        float di = dinv[i];
        float s = di * di;
        float4 hv = h[t];
        float4 bv = b[dq];
        float4 r;
        r.x = s * hv.x + bv.x;
        r.y = s * hv.y + bv.y;
        r.z = s * hv.z + bv.z;
        r.w = s * hv.w + bv.w;
        agg[t] = r;
    }
}

// ---------------------------------------------------------------------------
// edge scatter: one wave32 per edge, 3 coalesced dims per lane
// agg[col[e]][:] += dinv[row]*dinv[col] * h[row[e]][:]
// ---------------------------------------------------------------------------
__global__ void __launch_bounds__(256) k_scatter(
    const int* __restrict__ row, const int* __restrict__ col,
    const float* __restrict__ dinv, const float* __restrict__ h,
    float* __restrict__ agg) {
    int w = (blockIdx.x * blockDim.x + threadIdx.x) >> 5;
    int lane = threadIdx.x & 31;
    if (w >= N_EDGES) return;
    int r = row[w];
    int c = col[w];
    float nrm = dinv[r] * dinv[c];
    const float* hr = h + (size_t)r * HID_DIM;
    float* ac = agg + (size_t)c * HID_DIM;
#pragma unroll
    for (int j = 0; j < 3; ++j) {
        int d = lane + 32 * j;
        atomicAdd(ac + d, nrm * hr[d]);
    }
}

// ---------------------------------------------------------------------------
// global mean pool (sum phase): one wave per node
// ---------------------------------------------------------------------------
__global__ void __launch_bounds__(256) k_pool(
    const float* __restrict__ h, const int* __restrict__ batch,
    float* __restrict__ pooled, float* __restrict__ cnt) {
    int w = (blockIdx.x * blockDim.x + threadIdx.x) >> 5;
    int lane = threadIdx.x & 31;
    if (w >= N_NODES) return;
    int g = batch[w];
    const float* hr = h + (size_t)w * HID_DIM;
    float* pg = pooled + (size_t)g * HID_DIM;
#pragma unroll
    for (int j = 0; j < 3; ++j) {
        int d = lane + 32 * j;
        atomicAdd(pg + d, hr[d]);
    }
    if (lane == 0) atomicAdd(&cnt[g], 1.0f);
}

// ---------------------------------------------------------------------------
// classifier: logits = (pooled/cnt) @ Wf^T + bf ; log_softmax. 1 block/128 thr.
// ---------------------------------------------------------------------------
__global__ void k_final(const float* __restrict__ pooled,
                        const float* __restrict__ cnt,
                        const float* __restrict__ Wf,
                        const float* __restrict__ bf,
                        float* __restrict__ out) {
    int g = threadIdx.x;
    if (g >= N_GRAPHS) return;
    float inv = 1.0f / fmaxf(cnt[g], 1.0f);
    const float* pg = pooled + (size_t)g * HID_DIM;
    float logits[OUT_DIM];
#pragma unroll
    for (int o = 0; o < OUT_DIM; ++o) {
        float s = bf[o];
        const float* wo = Wf + (size_t)o * HID_DIM;
        for (int d = 0; d < HID_DIM; ++d) s += pg[d] * inv * wo[d];
        logits[o] = s;
    }
    float m = logits[0];
#pragma unroll
    for (int o = 1; o < OUT_DIM; ++o) m = fmaxf(m, logits[o]);
    float se = 0.0f;
#pragma unroll
    for (int o = 0; o < OUT_DIM; ++o) se += expf(logits[o] - m);
    float lse = m + logf(se);
#pragma unroll
    for (int o = 0; o < OUT_DIM; ++o) out[(size_t)g * OUT_DIM + o] = logits[o] - lse;
}

// ---------------------------------------------------------------------------
extern "C" void kernel_launch(void* const* d_in, const int* in_sizes, int n_in,
                              void* d_out, int out_size, void* d_ws, size_t ws_size,
                              hipStream_t stream) {
    const float* x   = (const float*)d_in[0];
    const int* eidx  = (const int*)d_in[1];
    const int* batch = (const int*)d_in[2];
    const float* W1  = (const float*)d_in[3];
    const float* b1  = (const float*)d_in[4];
    const float* W2  = (const float*)d_in[5];
    const float* b2  = (const float*)d_in[6];
    const float* Wf  = (const float*)d_in[7];
    const float* bf  = (const float*)d_in[8];
    float* out = (float*)d_out;

    const int* row = eidx;
    const int* col = eidx + N_EDGES;

    // workspace carve-up (floats)
    float* ws = (float*)d_ws;
    float* dinv   = ws;                                   // N
    float* bufA   = dinv + N_NODES;                       // N*96 (h after GEMM)
    float* bufB   = bufA + (size_t)N_NODES * HID_DIM;     // N*96 (aggregate)
    float* pooled = bufB + (size_t)N_NODES * HID_DIM;     // 128*96
    float* cnt    = pooled + N_GRAPHS * HID_DIM;          // 128
    int* deg = (int*)dinv;                                // aliased int phase

    const int numMTiles = (N_NODES + 15) / 16; // 3125 (exact)
    const int gemmBlocks = (numMTiles + 7) / 8;
    const int nq = N_NODES * (HID_DIM / 4);

    // 1) init counters
    k_init<<<(N_NODES + 255) / 256, 256, 0, stream>>>(deg, pooled, cnt);
    // 2) degree + dinv
    k_degree<<<(N_EDGES + 255) / 256, 256, 0, stream>>>(col, deg);
    k_dinv<<<(N_NODES + 255) / 256, 256, 0, stream>>>(deg, dinv);
    // 3) conv1: h1 = x @ W1 (WMMA), agg1 = selfloop+b1 then edge scatter
    k_gemm_wmma<IN_DIM, 0><<<gemmBlocks, 256, 0, stream>>>(x, W1, bufA,
                                                           N_NODES, numMTiles);
    k_selfloop<<<(nq + 255) / 256, 256, 0, stream>>>(
        (const float4*)bufA, dinv, (const float4*)b1, (float4*)bufB);
    k_scatter<<<(N_EDGES * 32 + 255) / 256, 256, 0, stream>>>(row, col, dinv,
                                                              bufA, bufB);
    // 4) conv2: h2 = relu(agg1) @ W2 (WMMA, ReLU fused into A load)
    k_gemm_wmma<HID_DIM, 1><<<gemmBlocks, 256, 0, stream>>>(bufB, W2, bufA,
                                                            N_NODES, numMTiles);
    k_selfloop<<<(nq + 255) / 256, 256, 0, stream>>>(
        (const float4*)bufA, dinv, (const float4*)b2, (float4*)bufB);
    k_scatter<<<(N_EDGES * 32 + 255) / 256, 256, 0, stream>>>(row, col, dinv,
                                                              bufA, bufB);
    // 5) mean pool + classifier + log_softmax
    k_pool<<<(N_NODES * 32 + 255) / 256, 256, 0, stream>>>(bufB, batch, pooled,
                                                           cnt);
    k_final<<<1, 128, 0, stream>>>(pooled, cnt, Wf, bf, out);
}